// NMS_44925357916696
// MI455X (gfx1250) — compile-verified
//
#include <hip/hip_runtime.h>
#include <stdint.h>

// ---------------- NMS configuration (from the reference) ----------------
#define NMS_CONF 0.5f
#define NMS_IOU  0.5f
#define NMS_NEG  (-1e9f)

#define TILE_T 1024   // threads per class-workgroup (32 wave32 waves)
#define VEC    4      // float4 box elements per thread per tile (tile = 4096 boxes)

// ---------------- gfx1250 async global->LDS path ----------------
#if defined(__HIP_DEVICE_COMPILE__)
  #if __has_builtin(__builtin_amdgcn_global_load_async_to_lds_b128)
    #define HAS_ASYNC_LDS 1
  #else
    #define HAS_ASYNC_LDS 0
  #endif
  #if __has_builtin(__builtin_amdgcn_s_wait_asynccnt)
    #define HAS_ASYNC_WAIT 1
  #else
    #define HAS_ASYNC_WAIT 0
  #endif
#else
  #define HAS_ASYNC_LDS 0
  #define HAS_ASYNC_WAIT 0
#endif

// Exact parameter type per hipcc diagnostic: pointer to
// `__attribute__((vector_size(4*sizeof(int)))) int` in AS1 (printed as
// `__device__` in HIP mode) / AS3 for the LDS side.
typedef int v4i_b128 __attribute__((vector_size(4 * sizeof(int))));
typedef __attribute__((address_space(1))) v4i_b128* as1_v4i;
typedef __attribute__((address_space(3))) v4i_b128* as3_v4i;

// 16-byte async copy global -> LDS. Falls back to a synchronous copy if the
// builtin is unavailable. AS1/AS3 pointers are built via integer round-trips:
// AS1 value == generic global address; AS3 value == low 32 bits of the
// generic LDS address (hardware truncates flat LDS addresses the same way).
__device__ __forceinline__ void lds_async_copy16(float4* lds_dst, const float4* gsrc) {
#if HAS_ASYNC_LDS
  __builtin_amdgcn_global_load_async_to_lds_b128(
      (as1_v4i)(uintptr_t)(const void*)gsrc,
      (as3_v4i)(uint32_t)(uintptr_t)(void*)lds_dst,
      0, 0);
#else
  *lds_dst = *gsrc;
#endif
}

template <int CNT>
__device__ __forceinline__ void lds_async_wait() {
#if HAS_ASYNC_LDS
  #if HAS_ASYNC_WAIT
  __builtin_amdgcn_s_wait_asynccnt(CNT);
  #else
  asm volatile("s_wait_asynccnt %0" :: "i"(CNT) : "memory");
  #endif
#endif
}

// ---------------- init kernels ----------------

// Transpose scores [N,C] -> ws [C,N] with confidence pre-threshold applied.
__global__ void nms_init_scores_t(const float* __restrict__ scores,
                                  float* __restrict__ ws, int N, int C) {
  int i = blockIdx.x * blockDim.x + threadIdx.x;
  if (i >= N) return;
  for (int c = 0; c < C; ++c) {
    float s = scores[(size_t)i * C + c];
    ws[(size_t)c * N + i] = (s >= NMS_CONF) ? s : NMS_NEG;
  }
}

__global__ void nms_zero_u32(unsigned int* __restrict__ p, size_t n) {
  size_t i = blockIdx.x * (size_t)blockDim.x + threadIdx.x;
  if (i < n) p[i] = 0u;
}

// ---------------- main kernel: one workgroup per class ----------------
//
// TRANSPOSED == true : scores live in ws_scores[C,N] (pre-thresholded);
//                      suppression writes NEG back in place (each element is
//                      owned by exactly one thread, so no races).
// TRANSPOSED == false: low-workspace fallback; suppression state is a bitmask
//                      ws_mask[C, N/32]; each wave32 owns its mask word and
//                      commits suppression via __ballot (no atomics needed).
//
// Per step k: one fused sweep over all N boxes applies suppression against
// the box selected at step k-1 (IoU(b,b)==1 also removes the selected box
// itself) while tracking the running argmax. Box tiles are double-buffered
// into LDS with async global->LDS copies; every thread reads only the LDS
// slots it issued, so the per-wave s_wait_asynccnt is the only sync needed
// inside the tile loop.
template <bool TRANSPOSED>
__global__ __launch_bounds__(TILE_T) void nms_per_class(
    const float4* __restrict__ boxes,
    const float*  __restrict__ raw_scores,   // [N,C], used when !TRANSPOSED
    float*        __restrict__ ws_scores,    // [C,N], used when TRANSPOSED
    unsigned int* __restrict__ ws_mask,      // [C,N/32], used when !TRANSPOSED
    float*        __restrict__ out,
    int N, int C, int K)
{
  const int cls = blockIdx.x;
  const int tid = threadIdx.x;

  __shared__ float4 tileBuf[2][TILE_T * VEC];  // 128 KB double buffer
  __shared__ float  red_s[TILE_T];
  __shared__ int    red_i[TILE_T];
  __shared__ float4 sBox;
  __shared__ int    sValid;

  float* sc = nullptr;
  unsigned int* mask = nullptr;
  if (TRANSPOSED) sc = ws_scores + (size_t)cls * N;
  else            mask = ws_mask + (size_t)cls * (size_t)((N + 31) / 32);

  const int tileElems = TILE_T * VEC;
  const int nTiles = (N + tileElems - 1) / tileElems;

  float* out_boxes  = out;                         // [C*K,4]
  float* out_scores = out + (size_t)C * K * 4;     // [C*K]
  float* out_class  = out + (size_t)C * K * 5;     // [C*K]
  float* out_valid  = out + (size_t)C * K * 6;     // [C*K]

  float4 lastBox = make_float4(0.f, 0.f, 0.f, 0.f);
  bool haveLast = false;                           // block-uniform

  for (int k = 0; k < K; ++k) {
    float bestS = NMS_NEG;
    int   bestI = 0;

    if (haveLast) {  // prefetch tile 0 (uniform branch)
      #pragma unroll
      for (int j = 0; j < VEC; ++j) {
        const int i = j * TILE_T + tid;
        if (i < N) lds_async_copy16(&tileBuf[0][j * TILE_T + tid], &boxes[i]);
      }
    }

    for (int t = 0; t < nTiles; ++t) {
      const int buf  = t & 1;
      const int base = t * tileElems;

      if (haveLast) {
        if (t + 1 < nTiles) {
          #pragma unroll
          for (int j = 0; j < VEC; ++j) {
            const int i = (t + 1) * tileElems + j * TILE_T + tid;
            if (i < N) lds_async_copy16(&tileBuf[buf ^ 1][j * TILE_T + tid], &boxes[i]);
          }
          lds_async_wait<VEC>();   // drain current tile, leave next in flight
        } else {
          lds_async_wait<0>();
        }
      }

      #pragma unroll
      for (int j = 0; j < VEC; ++j) {
        const int i = base + j * TILE_T + tid;
        float s = NMS_NEG;
        bool alive = false;
        if (i < N) {
          if (TRANSPOSED) {
            s = sc[i];
            alive = s > -1e8f;
          } else {
            const unsigned int m = mask[(unsigned)i >> 5];
            const float raw = raw_scores[(size_t)i * C + cls];
            alive = (raw >= NMS_CONF) && (((m >> (i & 31)) & 1u) == 0u);
            s = alive ? raw : NMS_NEG;
          }
        }

        bool supp = false;
        if (haveLast && alive) {
          const float4 b = tileBuf[buf][j * TILE_T + tid];
          const float yy1 = fmaxf(lastBox.x, b.x);
          const float xx1 = fmaxf(lastBox.y, b.y);
          const float yy2 = fminf(lastBox.z, b.z);
          const float xx2 = fminf(lastBox.w, b.w);
          const float inter = fmaxf(yy2 - yy1, 0.f) * fmaxf(xx2 - xx1, 0.f);
          const float a1 = (lastBox.z - lastBox.x) * (lastBox.w - lastBox.y);
          const float a2 = (b.z - b.x) * (b.w - b.y);
          const float iou = inter / (a1 + a2 - inter + 1e-9f);
          supp = iou > NMS_IOU;
        }

        if (supp) {
          s = NMS_NEG;
          if (TRANSPOSED) sc[i] = NMS_NEG;
        }
        if (!TRANSPOSED) {
          const unsigned int w = (unsigned int)__ballot(supp);  // wave32
          if (((tid & 31) == 0) && (w != 0u) && (i < N)) mask[(unsigned)i >> 5] |= w;
        }
        if (s > bestS) { bestS = s; bestI = i; }   // strict > keeps first max
      }
    }

    // Block-wide argmax (tie-break: lowest index, matching jnp.argmax).
    red_s[tid] = bestS;
    red_i[tid] = bestI;
    __syncthreads();
    for (int off = TILE_T / 2; off > 0; off >>= 1) {
      if (tid < off) {
        const float os = red_s[tid + off];
        const int   oi = red_i[tid + off];
        if (os > red_s[tid] || (os == red_s[tid] && oi < red_i[tid])) {
          red_s[tid] = os; red_i[tid] = oi;
        }
      }
      __syncthreads();
    }

    if (tid == 0) {
      const float sv = red_s[0];
      const int   iv = red_i[0];
      const int valid = (sv > 0.5f * NMS_NEG) ? 1 : 0;   // sc > -5e8
      const float4 b = boxes[iv];
      sBox = b;
      sValid = valid;
      const int slot = cls * K + k;
      ((float4*)out_boxes)[slot] = valid ? b : make_float4(0.f, 0.f, 0.f, 0.f);
      out_scores[slot] = valid ? sv : 0.f;
      out_class[slot]  = (float)cls;
      out_valid[slot]  = valid ? 1.f : 0.f;
    }
    __syncthreads();
    lastBox  = sBox;
    haveLast = (sValid != 0);
  }
}

// ---------------- host launcher ----------------
extern "C" void kernel_launch(void* const* d_in, const int* in_sizes, int n_in,
                              void* d_out, int out_size, void* d_ws, size_t ws_size,
                              hipStream_t stream) {
  const float* boxes  = (const float*)d_in[0];   // [N,4]
  const float* scores = (const float*)d_in[1];   // [N,C]

  const int N = in_sizes[0] / 4;
  const int C = (N > 0) ? (in_sizes[1] / N) : 1;
  int K = (C > 0) ? (out_size / (7 * C)) : 50;   // out = C*K*(4+1+1+1) floats
  if (K <= 0) K = 50;

  float* out = (float*)d_out;
  const size_t need_full = (size_t)C * (size_t)N * sizeof(float);
  const size_t mask_words = (size_t)C * (size_t)((N + 31) / 32);

  if (ws_size >= need_full) {
    // Primary path: pre-thresholded transposed scores in workspace (L2-resident).
    nms_init_scores_t<<<dim3((N + 255) / 256), dim3(256), 0, stream>>>(
        scores, (float*)d_ws, N, C);
    nms_per_class<true><<<dim3(C), dim3(TILE_T), 0, stream>>>(
        (const float4*)boxes, scores, (float*)d_ws, nullptr, out, N, C, K);
  } else {
    // Low-workspace fallback: per-class suppression bitmask + strided score reads.
    nms_zero_u32<<<dim3((unsigned)((mask_words + 255) / 256)), dim3(256), 0, stream>>>(
        (unsigned int*)d_ws, mask_words);
    nms_per_class<false><<<dim3(C), dim3(TILE_T), 0, stream>>>(
        (const float4*)boxes, scores, nullptr, (unsigned int*)d_ws, out, N, C, K);
  }
}